// GraphConv_78752520339637
// MI455X (gfx1250) — compile-verified
//
#include <hip/hip_runtime.h>

#define N_NODES 100000
#define D_IN    256
#define D_OUT   128
#define LN_EPS  1e-5f

typedef __attribute__((ext_vector_type(2))) float v2f;
typedef __attribute__((ext_vector_type(8))) float v8f;

// ---------------------------------------------------------------------------
// Kernel A: fused dual GEMM over a shared 16-row x tile.
//   waves 0..7  : support[:, w*16 : w*16+16] = x_tile @ weight  -> d_ws
//   waves 8..15 : out[:, w*16 : ...] = bias + relu(x_tile @ res_w + res_b)
// Uses V_WMMA_F32_16X16X4_F32 (native fp32 matrix op on CDNA5).
// ---------------------------------------------------------------------------
__global__ __launch_bounds__(512)
void gemm_init_kernel(const float* __restrict__ x,
                      const float* __restrict__ weight,
                      const float* __restrict__ res_w,
                      const float* __restrict__ bias,
                      const float* __restrict__ res_b,
                      float* __restrict__ support,
                      float* __restrict__ out)
{
    __shared__ __align__(16) float xs[16 * D_IN];   // 16 KB tile of x

    const int row0 = blockIdx.x * 16;               // 6250 row tiles exactly
    const int tid  = threadIdx.x;

    // 16x256 tile of row-major x is contiguous: 4096 floats = 1024 float4
    {
        const float4* s4 = (const float4*)(x + (size_t)row0 * D_IN);
        float4* d4 = (float4*)xs;
        d4[tid]       = s4[tid];
        d4[tid + 512] = s4[tid + 512];
    }
    __syncthreads();

    const int lane    = tid & 31;
    const int wave    = tid >> 5;        // 0..15
    const int colTile = wave & 7;        // 8 tiles cover D_OUT=128
    const int isRes   = wave >> 3;       // 0 = weight branch, 1 = res branch
    const float* __restrict__ W = isRes ? res_w : weight;

    // ISA 16x4 fp32 A layout: lanes 0-15 hold K=0,1; lanes 16-31 hold K=2,3.
    const int m    = lane & 15;
    const int n    = colTile * 16 + m;          // B/C/D column = lane%16
    const int kofs = (lane >> 4) << 1;          // 0 or 2

    v8f acc = {0.f, 0.f, 0.f, 0.f, 0.f, 0.f, 0.f, 0.f};

    #pragma unroll 8
    for (int k = 0; k < D_IN; k += 4) {
        v2f a, b;
        a.x = xs[m * D_IN + k + kofs];
        a.y = xs[m * D_IN + k + kofs + 1];
        b.x = W[(size_t)(k + kofs)     * D_OUT + n];
        b.y = W[(size_t)(k + kofs + 1) * D_OUT + n];
        acc = __builtin_amdgcn_wmma_f32_16x16x4_f32(
                  /*neg_a=*/false, a, /*neg_b=*/false, b,
                  /*c_mod=*/(short)0, acc, /*reuse_a=*/false, /*reuse_b=*/false);
    }

    // C/D layout: VGPR v -> row M=v (lanes 0-15) or M=v+8 (lanes 16-31)
    const int rbase = row0 + ((lane >> 4) << 3);
    if (isRes == 0) {
        #pragma unroll
        for (int v = 0; v < 8; ++v)
            support[(size_t)(rbase + v) * D_OUT + n] = acc[v];
    } else {
        const float bc  = bias[n];
        const float rbc = res_b[n];
        #pragma unroll
        for (int v = 0; v < 8; ++v) {
            float t = acc[v] + rbc;
            t = t > 0.f ? t : 0.f;
            out[(size_t)(rbase + v) * D_OUT + n] = bc + t;
        }
    }
}

// ---------------------------------------------------------------------------
// Kernel B: weighted COO scatter-add. One wave32 per edge, 4 floats per lane.
// support (51 MB) is L2-resident on MI455X (192 MB L2), so the random gather
// mostly hits L2. Atomics lower to global_atomic_add_f32.
// ---------------------------------------------------------------------------
__global__ __launch_bounds__(256)
void spmm_kernel(const float* __restrict__ support,
                 const float* __restrict__ edge_w,
                 const int*   __restrict__ edge_src,
                 const int*   __restrict__ edge_dst,
                 float* __restrict__ out,
                 int n_edges)
{
    const int gtid = blockIdx.x * 256 + threadIdx.x;
    const int edge = gtid >> 5;
    const int lane = threadIdx.x & 31;
    if (edge >= n_edges) return;

    const int   src = edge_src[edge];
    const int   dst = edge_dst[edge];
    const float w   = edge_w[edge];

    const float4 v = *(const float4*)(support + (size_t)src * D_OUT + lane * 4);
    float* o = out + (size_t)dst * D_OUT + lane * 4;

    __hip_atomic_fetch_add(o + 0, w * v.x, __ATOMIC_RELAXED, __HIP_MEMORY_SCOPE_AGENT);
    __hip_atomic_fetch_add(o + 1, w * v.y, __ATOMIC_RELAXED, __HIP_MEMORY_SCOPE_AGENT);
    __hip_atomic_fetch_add(o + 2, w * v.z, __ATOMIC_RELAXED, __HIP_MEMORY_SCOPE_AGENT);
    __hip_atomic_fetch_add(o + 3, w * v.w, __ATOMIC_RELAXED, __HIP_MEMORY_SCOPE_AGENT);
}

// ---------------------------------------------------------------------------
// Kernel C: in-place LayerNorm over D_OUT=128. One wave32 per node,
// 4 floats per lane, cross-lane reduction via shuffles (wave32!).
// ---------------------------------------------------------------------------
__global__ __launch_bounds__(256)
void ln_kernel(float* __restrict__ out,
               const float* __restrict__ gamma,
               const float* __restrict__ beta)
{
    const int gtid = blockIdx.x * 256 + threadIdx.x;
    const int node = gtid >> 5;
    const int lane = threadIdx.x & 31;
    if (node >= N_NODES) return;

    float4 v = *(float4*)(out + (size_t)node * D_OUT + lane * 4);
    float s  = v.x + v.y + v.z + v.w;
    float ss = v.x * v.x + v.y * v.y + v.z * v.z + v.w * v.w;

    #pragma unroll
    for (int off = 16; off > 0; off >>= 1) {
        s  += __shfl_xor(s,  off, 32);
        ss += __shfl_xor(ss, off, 32);
    }

    const float mu  = s * (1.0f / 128.0f);
    const float var = ss * (1.0f / 128.0f) - mu * mu;
    const float rs  = rsqrtf(var + LN_EPS);

    const float4 g = *(const float4*)(gamma + lane * 4);
    const float4 b = *(const float4*)(beta  + lane * 4);

    v.x = (v.x - mu) * rs * g.x + b.x;
    v.y = (v.y - mu) * rs * g.y + b.y;
    v.z = (v.z - mu) * rs * g.z + b.z;
    v.w = (v.w - mu) * rs * g.w + b.w;

    *(float4*)(out + (size_t)node * D_OUT + lane * 4) = v;
}

extern "C" void kernel_launch(void* const* d_in, const int* in_sizes, int n_in,
                              void* d_out, int out_size, void* d_ws, size_t ws_size,
                              hipStream_t stream)
{
    const float* x          = (const float*)d_in[0];
    const float* weight     = (const float*)d_in[1];
    const float* bias       = (const float*)d_in[2];
    const float* res_w      = (const float*)d_in[3];
    const float* res_b      = (const float*)d_in[4];
    const float* ln_gamma   = (const float*)d_in[5];
    const float* ln_beta    = (const float*)d_in[6];
    const float* edge_weight= (const float*)d_in[7];
    const int*   edge_index = (const int*)d_in[8];
    const int    n_edges    = in_sizes[7];

    float* out     = (float*)d_out;
    float* support = (float*)d_ws;   // 100000*128 floats = 51.2 MB scratch

    // 1) fused dual GEMM (WMMA fp32) + accumulator init
    gemm_init_kernel<<<N_NODES / 16, 512, 0, stream>>>(
        x, weight, res_w, bias, res_b, support, out);

    // 2) edge scatter-add
    const long long spmm_threads = (long long)n_edges * 32;
    const int spmm_blocks = (int)((spmm_threads + 255) / 256);
    spmm_kernel<<<spmm_blocks, 256, 0, stream>>>(
        support, edge_weight, edge_index, edge_index + n_edges, out, n_edges);

    // 3) layernorm in place
    const long long ln_threads = (long long)N_NODES * 32;
    const int ln_blocks = (int)((ln_threads + 255) / 256);
    ln_kernel<<<ln_blocks, 256, 0, stream>>>(out, ln_gamma, ln_beta);
}